// DataAwareGCN_80934363726496
// MI455X (gfx1250) — compile-verified
//
#include <hip/hip_runtime.h>
#include <hip/hip_bf16.h>

typedef __attribute__((ext_vector_type(2))) float v2f;
typedef __attribute__((ext_vector_type(8))) float v8f;

#define IN_DIM 128
#define HID 64
#define OUTD 32

// ---------------------------------------------------------------------------
// fill: set n floats to val
// ---------------------------------------------------------------------------
__global__ void k_fill(float* __restrict__ p, float val, int n) {
  int t = blockIdx.x * blockDim.x + threadIdx.x;
  if (t < n) p[t] = val;
}

// ---------------------------------------------------------------------------
// degree accumulation: deg[dst[e]] += 1  (deg pre-filled with 1.0 = self loop)
// ---------------------------------------------------------------------------
__global__ void k_deg(const int* __restrict__ dst, float* __restrict__ deg, int E) {
  int e = blockIdx.x * blockDim.x + threadIdx.x;
  if (e < E) atomicAdd(&deg[dst[e]], 1.0f);
}

__global__ void k_dinv(const float* __restrict__ deg, float* __restrict__ dinv, int n) {
  int t = blockIdx.x * blockDim.x + threadIdx.x;
  if (t < n) dinv[t] = rsqrtf(deg[t]);
}

// ---------------------------------------------------------------------------
// Dense transform via V_WMMA_F32_16X16X4_F32, fused with per-row dinv scale:
//   out[m, n] = dinv[m] * sum_k A[m,k] * W[k,n]
// One wave computes one 16x16 output tile. Lane layouts per CDNA5 ISA 7.12.2:
//   A frag (v2f): lane l, vgpr j -> A[l&15][2*(l>>4) + j]
//   B frag (v2f): lane l, vgpr j -> W[2*(l>>4) + j][l&15]
//   D frag (v8f): vgpr r, lane l -> D[r + 8*(l>>4)][l&15]
// numTiles waves total; guard is wave-uniform so EXEC is all-ones for WMMA.
// ---------------------------------------------------------------------------
template <int K, int NOUT>
__global__ void k_gemm_wmma(const float* __restrict__ A, const float* __restrict__ W,
                            const float* __restrict__ dinv, float* __restrict__ out,
                            int numTiles) {
  constexpr int NT = NOUT / 16;
  int wave = (blockIdx.x * blockDim.x + threadIdx.x) >> 5;
  if (wave >= numTiles) return;
  int lane = threadIdx.x & 31;
  int nt = wave % NT;
  int mt = wave / NT;
  int half = lane >> 4;
  int l16 = lane & 15;
  int m = mt * 16 + l16;
  int n = nt * 16 + l16;
  int kh = half * 2;

  const float* arow = A + (size_t)m * K;
  v8f acc = {};
#pragma unroll
  for (int k0 = 0; k0 < K; k0 += 4) {
    // A: two consecutive K elements of row m -> 8B vector load
    float2 av = *(const float2*)(arow + k0 + kh);
    v2f a;
    a[0] = av.x;
    a[1] = av.y;
    // B: two rows of W at column n (W is tiny, L0/L2 resident)
    v2f b;
    b[0] = W[(size_t)(k0 + kh) * NOUT + n];
    b[1] = W[(size_t)(k0 + kh + 1) * NOUT + n];
    acc = __builtin_amdgcn_wmma_f32_16x16x4_f32(
        /*neg_a=*/false, a, /*neg_b=*/false, b,
        /*c_mod=*/(short)0, acc, /*reuse_a=*/false, /*reuse_b=*/false);
  }
#pragma unroll
  for (int r = 0; r < 8; ++r) {
    int row = mt * 16 + r + 8 * half;
    out[(size_t)row * NOUT + nt * 16 + l16] = acc[r] * dinv[row];
  }
}

// ---------------------------------------------------------------------------
// Edge scatter: agg[dst[e]*F + f] += hs[src[e]*F + f]
// F is 32 or 64 -> edge index is wave-uniform; readfirstlane scalarizes the
// src/dst loads (s_load instead of 32 redundant vmem lanes).
// ---------------------------------------------------------------------------
template <int F>
__global__ void k_scatter(const float* __restrict__ hs, const int* __restrict__ src,
                          const int* __restrict__ dst, float* __restrict__ agg,
                          unsigned total) {
  unsigned t = blockIdx.x * blockDim.x + threadIdx.x;
  if (t >= total) return;
  int f = (int)(t & (F - 1));
  int e = (int)(t / F);
  e = __builtin_amdgcn_readfirstlane(e);
  int s = src[e];
  int d = dst[e];
  atomicAdd(&agg[(size_t)d * F + f], hs[(size_t)s * F + f]);
}

// ---------------------------------------------------------------------------
// Combine: out = relu(dinv[i] * (agg + hs_selfloop) + b[f])
// ---------------------------------------------------------------------------
template <int F>
__global__ void k_combine(const float* __restrict__ agg, const float* __restrict__ hs,
                          const float* __restrict__ dinv, const float* __restrict__ b,
                          float* __restrict__ out, int n) {
  int t = blockIdx.x * blockDim.x + threadIdx.x;
  if (t >= n) return;
  int i = t / F;
  int f = t & (F - 1);
  float v = dinv[i] * (agg[t] + hs[t]) + b[f];
  out[t] = fmaxf(v, 0.0f);
}

// ---------------------------------------------------------------------------
extern "C" void kernel_launch(void* const* d_in, const int* in_sizes, int n_in,
                              void* d_out, int out_size, void* d_ws, size_t ws_size,
                              hipStream_t stream) {
  const float* x  = (const float*)d_in[0];
  const int*   ei = (const int*)d_in[1];
  const float* W1 = (const float*)d_in[2];
  const float* b1 = (const float*)d_in[3];
  const float* W2 = (const float*)d_in[4];
  const float* b2 = (const float*)d_in[5];
  float* out = (float*)d_out;

  const int N = in_sizes[0] / IN_DIM;   // 100000
  const int E = in_sizes[1] / 2;        // 1600000
  const int* src = ei;
  const int* dst = ei + E;

  // Workspace layout (floats): deg | dinv | hs1 | agg1(->h1) | hs2 | agg2
  float* ws   = (float*)d_ws;
  float* deg  = ws;
  float* dinv = deg + N;
  float* hs1  = dinv + N;
  float* agg1 = hs1 + (size_t)N * HID;
  float* hs2  = agg1 + (size_t)N * HID;
  float* agg2 = hs2 + (size_t)N * OUTD;

  const int B = 256;
  auto blk = [](long long n, int b) { return (unsigned)((n + b - 1) / b); };

  // init
  k_fill<<<blk(N, B), B, 0, stream>>>(deg, 1.0f, N);
  k_fill<<<blk((long long)N * HID, B), B, 0, stream>>>(agg1, 0.0f, N * HID);
  k_fill<<<blk((long long)N * OUTD, B), B, 0, stream>>>(agg2, 0.0f, N * OUTD);

  // degree + normalization
  k_deg<<<blk(E, B), B, 0, stream>>>(dst, deg, E);
  k_dinv<<<blk(N, B), B, 0, stream>>>(deg, dinv, N);

  // layer 1: hs1 = (x @ W1) * dinv[row]
  {
    int tiles = (N / 16) * (HID / 16);  // 6250 * 4 = 25000 waves
    k_gemm_wmma<IN_DIM, HID><<<blk((long long)tiles * 32, B), B, 0, stream>>>(
        x, W1, dinv, hs1, tiles);
  }
  // edge aggregation
  {
    unsigned total = (unsigned)E * HID;  // 102.4M
    k_scatter<HID><<<blk(total, B), B, 0, stream>>>(hs1, src, dst, agg1, total);
  }
  // h1 = relu(dinv*(agg1 + hs1) + b1), written in place into agg1
  k_combine<HID><<<blk((long long)N * HID, B), B, 0, stream>>>(
      agg1, hs1, dinv, b1, agg1, N * HID);

  // layer 2: hs2 = (h1 @ W2) * dinv[row]
  {
    int tiles = (N / 16) * (OUTD / 16);  // 6250 * 2 = 12500 waves
    k_gemm_wmma<HID, OUTD><<<blk((long long)tiles * 32, B), B, 0, stream>>>(
        agg1, W2, dinv, hs2, tiles);
  }
  {
    unsigned total = (unsigned)E * OUTD;  // 51.2M
    k_scatter<OUTD><<<blk(total, B), B, 0, stream>>>(hs2, src, dst, agg2, total);
  }
  // out = relu(dinv*(agg2 + hs2) + b2)
  k_combine<OUTD><<<blk((long long)N * OUTD, B), B, 0, stream>>>(
      agg2, hs2, dinv, b2, out, N * OUTD);
}